// UsmFilter_33672543601411
// MI455X (gfx1250) — compile-verified
//
#include <hip/hip_runtime.h>
#include <math.h>

// MI455X / gfx1250, wave32. Fused separable Gaussian unsharp-mask:
// one HBM read + one HBM write (~100 MB @ 23.3 TB/s ~ 4.3 us floor),
// both 25-tap 1-D convolutions executed as banded matmuls on
// V_WMMA_F32_16X16X4_F32 through LDS.

typedef float v2f __attribute__((ext_vector_type(2)));
typedef float v8f __attribute__((ext_vector_type(8)));

#define IMG_H 512
#define IMG_W 512
#define RAD   12
#define KS    25           // 2*RAD+1 taps
#define WAVES 4            // waves per block (wave32)
#define IN_ROWS 48         // 16 output rows + 2*RAD halo + rg2 overshoot
#define IN_COLS 88         // 64 output cols + 2*RAD halo
#define PITCH_IN 90        // padded LDS pitch (floats), keeps b64 alignment
#define PITCH_HB 68

__device__ __forceinline__ int refl(int i, int n) {
    // jnp.pad(mode="reflect"): -1 -> 1, n -> n-2 (radius << n, single fold)
    i = (i < 0) ? -i : i;
    return (i >= n) ? (2 * n - 2 - i) : i;
}

__global__ __launch_bounds__(128) void usm_wmma_kernel(
        const float* __restrict__ img,
        const float* __restrict__ param,
        float* __restrict__ out)
{
    __shared__ float s_in[IN_ROWS * PITCH_IN];   // input halo tile
    __shared__ float s_hb[IN_ROWS * PITCH_HB];   // horizontally blurred tile
    __shared__ float s_g[32];                    // normalized 1-D Gaussian
    __shared__ float s_p;                        // unsharp strength for batch

    const int tid   = threadIdx.x;
    const int plane = blockIdx.z;            // b*3 + c, 0..47
    const int r0    = blockIdx.y * 16;       // output row base
    const int c0    = blockIdx.x * 64;       // output col base (block)

    // --- normalized Gaussian taps (runtime, matches reference formula) ---
    if (tid < KS) {
        float x = (float)(tid - RAD) * 0.2f;       // /sigma, sigma=5
        float w = expf(-0.5f * x * x);
        float s = 0.0f;
        for (int t = 0; t < KS; ++t) {
            float xt = (float)(t - RAD) * 0.2f;
            s += expf(-0.5f * xt * xt);
        }
        s_g[tid] = w / s;
    }
    if (tid == 0) {
        float pr = param[plane / 3];
        s_p = (tanhf(pr) * 0.5f + 0.5f) * 5.0f;    // tanh_range(0,5)
    }

    // --- cooperative reflect-indexed halo load: 48x88 tile -> LDS ---
    const float* src = img + (size_t)plane * (IMG_H * IMG_W);
    for (int idx = tid; idx < IN_ROWS * IN_COLS; idx += 128) {
        int tr = idx / IN_COLS;
        int tc = idx - tr * IN_COLS;
        int gr = refl(r0 - RAD + tr, IMG_H);
        int gc = refl(c0 - RAD + tc, IMG_W);
        s_in[tr * PITCH_IN + tc] = src[gr * IMG_W + gc];
    }
    __syncthreads();

    const int lane = tid & 31;
    const int wave = tid >> 5;
    const int half = lane >> 4;     // lane half selects K pair / row group
    const int l16  = lane & 15;
    const int woff = wave * 16;     // this wave's column slice in the block

    // ---------------- horizontal pass: D = In(16x40) x Wband(40x16) -------
    // B frag (banded Gaussian, const per lane): B[jj][n] = g[jj-n] if 0<=jj-n<25
    v2f wfrag[10];
#pragma unroll
    for (int k = 0; k < 10; ++k) {
        int jj = 4 * k + 2 * half;            // lanes>=16 hold K rows 2,3
        int d0 = jj - l16, d1 = d0 + 1;
        wfrag[k].x = (d0 >= 0 && d0 < KS) ? s_g[d0] : 0.0f;
        wfrag[k].y = (d1 >= 0 && d1 < KS) ? s_g[d1] : 0.0f;
    }

#pragma unroll
    for (int rg = 0; rg < 3; ++rg) {          // hblur rows r0-12 .. r0+35
        v8f acc = {};
        const float* arp = &s_in[(rg * 16 + l16) * PITCH_IN + woff + 2 * half];
#pragma unroll
        for (int k = 0; k < 10; ++k) {
            v2f a;
            a.x = arp[4 * k + 0];
            a.y = arp[4 * k + 1];
            acc = __builtin_amdgcn_wmma_f32_16x16x4_f32(
                    false, a, false, wfrag[k], (short)0, acc, false, false);
        }
        // D layout: VGPR v -> rows v (lanes 0-15) and v+8 (lanes 16-31)
#pragma unroll
        for (int v = 0; v < 8; ++v)
            s_hb[(rg * 16 + v + 8 * half) * PITCH_HB + woff + l16] = acc[v];
    }
    __syncthreads();

    // ---------------- vertical pass: D = Gband(16x40) x Hblur(40x16) ------
    v2f gfrag[10];
#pragma unroll
    for (int k = 0; k < 10; ++k) {
        int kk = 4 * k + 2 * half;            // A: lanes>=16 hold K cols 2,3
        int d0 = kk - l16, d1 = d0 + 1;       // G[m][kk] = g[kk-m] banded
        gfrag[k].x = (d0 >= 0 && d0 < KS) ? s_g[d0] : 0.0f;
        gfrag[k].y = (d1 >= 0 && d1 < KS) ? s_g[d1] : 0.0f;
    }

    v8f acc = {};
    const int col = woff + l16;
#pragma unroll
    for (int k = 0; k < 10; ++k) {
        int row = 4 * k + 2 * half;           // B rows K=0,1 / K=2,3 per half
        v2f b;
        b.x = s_hb[(row + 0) * PITCH_HB + col];
        b.y = s_hb[(row + 1) * PITCH_HB + col];
        acc = __builtin_amdgcn_wmma_f32_16x16x4_f32(
                false, gfrag[k], false, b, (short)0, acc, false, false);
    }

    // ---------------- unsharp combine + clip + coalesced store ------------
    const float p = s_p;
    float* dst = out + (size_t)plane * (IMG_H * IMG_W);
#pragma unroll
    for (int v = 0; v < 8; ++v) {
        int m = v + 8 * half;                               // output row in tile
        float orig = s_in[(RAD + m) * PITCH_IN + RAD + col]; // img from LDS
        float val  = (orig - acc[v]) * p + orig;
        val = fminf(fmaxf(val, 0.0f), 1.0f);
        dst[(r0 + m) * IMG_W + (c0 + col)] = val;
    }
}

extern "C" void kernel_launch(void* const* d_in, const int* in_sizes, int n_in,
                              void* d_out, int out_size, void* d_ws, size_t ws_size,
                              hipStream_t stream) {
    (void)in_sizes; (void)n_in; (void)out_size; (void)d_ws; (void)ws_size;
    const float* img   = (const float*)d_in[0];   // [16,3,512,512] f32
    const float* param = (const float*)d_in[1];   // [16] f32
    float* out = (float*)d_out;                   // [16,3,512,512] f32

    dim3 grid(IMG_W / 64, IMG_H / 16, 16 * 3);    // 8 x 32 x 48 blocks
    usm_wmma_kernel<<<grid, 128, 0, stream>>>(img, param, out);
}